// SSTInputLayerV2_18382460027103
// MI455X (gfx1250) — compile-verified
//
#include <hip/hip_runtime.h>
#include <hip/hip_bf16.h>
#include <stdint.h>
#include <stddef.h>

// ---------------------------------------------------------------------------
// SST input layer pipeline for gfx1250 (MI455X), wave32.
// Problem constants: WINDOW 12x12x1 over SPARSE 468x468x1 -> mnx=mny=40,
// mnz=2, mper=3200; batch<4 so window ids < 12800; flat2win composite key
// lvl*12800+bwi < 38400 < 2^16  => 2-pass 8-bit stable LSD radix sorts.
// ---------------------------------------------------------------------------

#define MPER    3200
#define NBINS_W 12800
#define NBINS_F 38400

typedef float v2f __attribute__((ext_vector_type(2)));
typedef float v8f __attribute__((ext_vector_type(8)));

// ------------------------------ tiny utilities ------------------------------

__global__ void k_init(int* scal, int n, int khost) {
  if (threadIdx.x == 0) { scal[0] = n; scal[1] = 0; scal[2] = 0; scal[3] = khost; }
}

__global__ void k_zero(int* p, int L) {
  int i = blockIdx.x * blockDim.x + threadIdx.x;
  if (i < L) p[i] = 0;
}

// ------------------------------ window coords -------------------------------

__global__ void k_windows(const int* __restrict__ coors, int n,
                          int* __restrict__ bwi0, int* __restrict__ bwi1) {
  int i = blockIdx.x * blockDim.x + threadIdx.x;
  if (i >= n) return;
  int b = coors[4 * i + 0];
  int y = coors[4 * i + 2];
  int x = coors[4 * i + 3];
  // z == 0 always; wz == sz so shz == 0 for both shifts -> wcz == 0.
  int cx0 = x + 12, cy0 = y + 12;                 // shift = false: sh = window
  bwi0[i] = b * MPER + (cx0 / 12) * 80 + (cy0 / 12) * 2;
  int cx1 = x + 6, cy1 = y + 6;                   // shift = true: sh = window/2
  bwi1[i] = b * MPER + (cx1 / 12) * 80 + (cy1 / 12) * 2;
}

__global__ void k_prepsort(const int* __restrict__ src, int L,
                           int* __restrict__ kA, int* __restrict__ iA) {
  int i = blockIdx.x * blockDim.x + threadIdx.x;
  if (i < L) { kA[i] = src[i]; iA[i] = i; }
}

// ------------------------- stable radix sort (8-bit) ------------------------

__global__ void k_hist(const int* __restrict__ keys, const int* dN, int shift,
                       int* __restrict__ gh, int NB) {
  __shared__ int h[256];
  int t = threadIdx.x;
  h[t] = 0;
  __syncthreads();
  int bound = dN[0];
  int i = blockIdx.x * 256 + t;
  if (i < bound) atomicAdd(&h[(keys[i] >> shift) & 255], 1);
  __syncthreads();
  gh[t * NB + blockIdx.x] = h[t];                 // digit-major for the scan
}

__global__ void k_scatter(const int* __restrict__ kin, const int* __restrict__ vin,
                          const int* dN, int shift, const int* __restrict__ ghs,
                          int NB, int* __restrict__ kout, int* __restrict__ vout) {
  __shared__ int dig[256];
  int t = threadIdx.x;
  int bound = dN[0];
  int i = blockIdx.x * 256 + t;
  int key = 0, val = 0, d = -1;
  if (i < bound) { key = kin[i]; val = vin[i]; d = (key >> shift) & 255; }
  dig[t] = d;
  __syncthreads();
  if (i < bound) {
    int r = 0;
    for (int j = 0; j < t; ++j) r += (dig[j] == d) ? 1 : 0;   // stable in-block rank
    int pos = ghs[d * NB + blockIdx.x] + r;
    kout[pos] = key;
    vout[pos] = val;
  }
}

// ------------------------------ exclusive scan ------------------------------

__global__ void k_scan_block(const int* __restrict__ in, int* __restrict__ out,
                             int* __restrict__ bsums, int L) {
  __shared__ int s[256];
  int t = threadIdx.x;
  int i = blockIdx.x * 256 + t;
  int x = (i < L) ? in[i] : 0;
  s[t] = x;
  __syncthreads();
  for (int off = 1; off < 256; off <<= 1) {
    int v = (t >= off) ? s[t - off] : 0;
    __syncthreads();
    s[t] += v;
    __syncthreads();
  }
  if (i < L) out[i] = s[t] - x;
  if (t == 255) bsums[blockIdx.x] = s[255];
}

__global__ void k_scan_top(int* __restrict__ bsums, int nb, int* total) {
  __shared__ int s[1024];
  int t = threadIdx.x;
  int x = (t < nb) ? bsums[t] : 0;
  s[t] = x;
  __syncthreads();
  for (int off = 1; off < 1024; off <<= 1) {
    int v = (t >= off) ? s[t - off] : 0;
    __syncthreads();
    s[t] += v;
    __syncthreads();
  }
  if (t < nb) bsums[t] = s[t] - x;
  if (t == 1023 && total != nullptr) total[0] = s[1023];
}

__global__ void k_scan_add(int* __restrict__ out, const int* __restrict__ bsums, int L) {
  int i = blockIdx.x * 256 + threadIdx.x;
  if (i < L) out[i] += bsums[blockIdx.x];
}

// --------------------------- per-window statistics --------------------------

__global__ void k_histfull(const int* __restrict__ keys, const int* dN,
                           int* __restrict__ cnts) {
  int i = blockIdx.x * blockDim.x + threadIdx.x;
  if (i < dN[0]) atomicAdd(&cnts[keys[i]], 1);
}

__global__ void k_inner_drop(const int* __restrict__ sk, const int* __restrict__ si,
                             const int* dN, const int* __restrict__ offs,
                             const int* __restrict__ cnts,
                             int* __restrict__ keep, int* __restrict__ lvlA) {
  int i = blockIdx.x * blockDim.x + threadIdx.x;
  if (i >= dN[0]) return;
  int k = sk[i];
  int inner = i - offs[k];                        // stable rank inside window
  int c = cnts[k];
  int target = 0, lvl = -1;
  if (c < 30)          { target = 30;  lvl = 0; }
  else if (c < 60)     { target = 60;  lvl = 1; }
  else if (c < 100000) { target = 100; lvl = 2; }
  int o = si[i];
  keep[o] = (inner < target) ? 1 : 0;
  lvlA[o] = lvl;
}

__global__ void k_compact0(const int* __restrict__ keep, const int* __restrict__ pos,
                           const int* __restrict__ bwi1, int n,
                           int* __restrict__ k0list, int* __restrict__ bwi1k) {
  int i = blockIdx.x * blockDim.x + threadIdx.x;
  if (i < n && keep[i]) { int p = pos[i]; k0list[p] = i; bwi1k[p] = bwi1[i]; }
}

__global__ void k_compact1(const int* __restrict__ keep, const int* __restrict__ pos,
                           const int* __restrict__ k0list, const int* __restrict__ lvlT,
                           const int* dK0, int* __restrict__ keepi, int* __restrict__ lvl1k) {
  int j = blockIdx.x * blockDim.x + threadIdx.x;
  if (j < dK0[0] && keep[j]) { int p = pos[j]; keepi[p] = k0list[j]; lvl1k[p] = lvlT[j]; }
}

__global__ void k_gather_kept(const int* __restrict__ keepi, const int* __restrict__ bwi0,
                              const int* __restrict__ bwi1, const int* __restrict__ lvl0,
                              int K, int* __restrict__ bwi0k, int* __restrict__ bwi1k2,
                              int* __restrict__ lvl0k) {
  int j = blockIdx.x * blockDim.x + threadIdx.x;
  if (j >= K) return;
  int r = keepi[j];
  bwi0k[j] = bwi0[r];
  bwi1k2[j] = bwi1[r];
  lvl0k[j] = lvl0[r];
}

// ------------------------------- flat2win ids -------------------------------

__global__ void k_makekey(const int* __restrict__ bwiK, const int* __restrict__ lvlK,
                          int K, int* __restrict__ keyF) {
  int j = blockIdx.x * blockDim.x + threadIdx.x;
  if (j < K) keyF[j] = lvlK[j] * NBINS_W + bwiK[j];
}

__global__ void k_used(const int* __restrict__ cnts, int L, int* __restrict__ used) {
  int i = blockIdx.x * blockDim.x + threadIdx.x;
  if (i < L) used[i] = (cnts[i] > 0) ? 1 : 0;
}

__global__ void k_f2w(const int* __restrict__ sk, const int* __restrict__ si, int K,
                      const int* __restrict__ offs, const int* __restrict__ cum,
                      int* __restrict__ fout) {
  int i = blockIdx.x * blockDim.x + threadIdx.x;
  if (i >= K) return;
  int k = sk[i];
  int inner = i - offs[k];
  int lvl = k / NBINS_W;
  int conti = cum[k] - cum[lvl * NBINS_W];        // unique-rank within level
  int mt = (lvl == 0) ? 30 : ((lvl == 1) ? 60 : 100);
  fout[si[i]] = conti * mt + inner;
}

// ------------------- feature gather via async-to-LDS (CDNA5) ----------------

__global__ void k_out_feats(const float* __restrict__ feats, const int* __restrict__ keepi,
                            float* __restrict__ o, int K) {
  __shared__ float4 buf[256];
  int t = threadIdx.x;
  int rl = t >> 5;                                // 8 rows / block
  int c4 = t & 31;                                // 32 float4 = 128 floats / row
  int row = blockIdx.x * 8 + rl;
  int src = (row < K) ? keepi[row] : 0;
  const float4* g = reinterpret_cast<const float4*>(feats) + (size_t)src * 32 + c4;
  unsigned loff = (unsigned)(uintptr_t)(&buf[t]);
  unsigned long long ga = (unsigned long long)(uintptr_t)g;
  asm volatile("global_load_async_to_lds_b128 %0, %1, off"
               :: "v"(loff), "v"(ga) : "memory");
  asm volatile("s_wait_asynccnt 0" ::: "memory");
  __syncthreads();
  if (row < K) reinterpret_cast<float4*>(o)[(size_t)row * 32 + c4] = buf[t];
}

// ------------------------------ misc outputs --------------------------------

__global__ void k_out_misc(const int* __restrict__ coors, const int* __restrict__ keepi,
                           const int* __restrict__ bwi0k, const int* __restrict__ bwi1k2,
                           const int* __restrict__ lvl0k, const int* __restrict__ lvl1k,
                           const int* __restrict__ f0i, const int* __restrict__ f1i,
                           float* o_coors, float* o_keep, float* o_b0, float* o_b1,
                           float* o_l0, float* o_l1, float* o_f0, float* o_f1, int K) {
  int j = blockIdx.x * blockDim.x + threadIdx.x;
  if (j >= K) return;
  int r = keepi[j];
  o_coors[4 * j + 0] = (float)coors[4 * r + 0];
  o_coors[4 * j + 1] = (float)coors[4 * r + 1];
  o_coors[4 * j + 2] = (float)coors[4 * r + 2];
  o_coors[4 * j + 3] = (float)coors[4 * r + 3];
  o_keep[j] = (float)r;
  o_b0[j] = (float)bwi0k[j];
  o_b1[j] = (float)bwi1k2[j];
  o_l0[j] = (float)lvl0k[j];
  o_l1[j] = (float)lvl1k[j];
  o_f0[j] = (float)f0i[j];
  o_f1[j] = (float)f1i[j];
}

// ----------------- positional embedding via V_WMMA_F32_16X16X4 --------------
// Phase matrix P[m][n] = v_m * (1/f_n) is a rank-1 16x4 * 4x16 f32 matmul:
// A col0 = per-voxel coordinate, B row0 = reciprocal frequencies, rest zero.
// One wave handles 16 voxels; fully unrolled -> 8 back-to-back v_wmma ops
// covering {shift0,shift1} x {x,y} x {freqs 0-15, 16-31}.
// Phases are bounded (|v|<=6, rf<=1) so hardware v_sin_f32/v_cos_f32 via
// __sinf/__cosf are exact-enough and co-execute with VALU as TRANS ops.

__global__ __launch_bounds__(32) void k_pos(const int* __restrict__ coors,
                                            const int* __restrict__ keepi,
                                            float* __restrict__ o_pos0,
                                            float* __restrict__ o_pos1, int K) {
  const int lane = threadIdx.x;
  const int base = blockIdx.x * 16;
  float vv[2][2] = {{0.f, 0.f}, {0.f, 0.f}};      // [shift][axis]
  if (lane < 16 && base + lane < K) {
    int r = keepi[base + lane];
    int y = coors[4 * r + 2], x = coors[4 * r + 3];
    vv[0][0] = (float)((x + 12) % 12) - 6.0f;     // ciw0.x - wx/2
    vv[0][1] = (float)((y + 12) % 12) - 6.0f;
    vv[1][0] = (float)((x + 6) % 12) - 6.0f;      // ciw1
    vv[1][1] = (float)((y + 6) % 12) - 6.0f;
  }
  const int n = lane & 15;
  const int half = lane >> 4;
  const float kLn = 0.28782313662425572f;          // ln(10000)/32
  // reciprocal frequencies for the two 16-wide halves (computed once)
  float rfh[2];
  rfh[0] = (lane < 16) ? __expf(-(float)n * kLn) : 0.f;
  rfh[1] = (lane < 16) ? __expf(-(float)(16 + n) * kLn) : 0.f;

#pragma unroll
  for (int s = 0; s < 2; ++s) {
    float* o = s ? o_pos1 : o_pos0;
#pragma unroll
    for (int axis = 0; axis < 2; ++axis) {
      float v = vv[s][axis];
#pragma unroll
      for (int h = 0; h < 2; ++h) {
        v2f a;  a.x = (lane < 16) ? v : 0.f;  a.y = 0.f;   // A: 16x4, only K=0
        v2f bb; bb.x = rfh[h];                bb.y = 0.f;  // B: 4x16, only K=0
        v8f c = {0.f, 0.f, 0.f, 0.f, 0.f, 0.f, 0.f, 0.f};
        v8f d = __builtin_amdgcn_wmma_f32_16x16x4_f32(
            false, a, false, bb, (short)0, c, false, false);
#pragma unroll
        for (int k = 0; k < 8; ++k) {
          int m = half * 8 + k;                   // C/D layout: VGPR k -> M=k / k+8
          int row = base + m;
          if (row < K) {
            float ph = d[k];                      // v_m * rf_n
            float2 sc;
            sc.x = __sinf(ph);                    // v_sin_f32
            sc.y = __cosf(ph);                    // v_cos_f32
            size_t off = (size_t)row * 128 + axis * 64 + 2 * (h * 16 + n);
            *reinterpret_cast<float2*>(o + off) = sc;   // one b64 store
          }
        }
      }
    }
  }
}

// ------------------------------- host helpers -------------------------------

static void run_scan(const int* in, int* out, int* bs, int L, int* total,
                     hipStream_t s) {
  int nb = (L + 255) / 256;
  k_scan_block<<<nb, 256, 0, s>>>(in, out, bs, L);
  k_scan_top<<<1, 1024, 0, s>>>(bs, nb, total);
  k_scan_add<<<nb, 256, 0, s>>>(out, bs, L);
}

static void radix_sort16(int* kA, int* iA, int* kB, int* iB, const int* dN,
                         int* gh, int* ghs, int* bs, int NB, hipStream_t s) {
  k_hist<<<NB, 256, 0, s>>>(kA, dN, 0, gh, NB);
  run_scan(gh, ghs, bs, 256 * NB, nullptr, s);
  k_scatter<<<NB, 256, 0, s>>>(kA, iA, dN, 0, ghs, NB, kB, iB);
  k_hist<<<NB, 256, 0, s>>>(kB, dN, 8, gh, NB);
  run_scan(gh, ghs, bs, 256 * NB, nullptr, s);
  k_scatter<<<NB, 256, 0, s>>>(kB, iB, dN, 8, ghs, NB, kA, iA);
}

extern "C" void kernel_launch(void* const* d_in, const int* in_sizes, int n_in,
                              void* d_out, int out_size, void* d_ws, size_t ws_size,
                              hipStream_t stream) {
  (void)n_in; (void)ws_size;
  const float* feats = (const float*)d_in[0];
  const int* coors = (const int*)d_in[1];
  const int n = in_sizes[1] / 4;                  // N_VOXELS
  const int K = out_size / 395;                   // 128+4+7+128+128 per kept voxel
  const int NB = (n + 255) / 256;
  const int GH = 256 * NB;
  const int KB = (K + 255) / 256;

  // ---- bump allocator over d_ws ----
  char* w = (char*)d_ws;
  auto alloc = [&](size_t bytes) -> int* {
    int* p = (int*)w;
    w += (bytes + 255) & ~(size_t)255;
    return p;
  };
  int* scal   = alloc(64);                 // [0]=n [1]=K0 [2]=Kdev [3]=Khost
  int* bwi0   = alloc((size_t)n * 4);
  int* bwi1   = alloc((size_t)n * 4);
  int* lvl0   = alloc((size_t)n * 4);
  int* kA     = alloc((size_t)n * 4);
  int* iA     = alloc((size_t)n * 4);
  int* kB     = alloc((size_t)n * 4);
  int* iB     = alloc((size_t)n * 4);
  int* gh     = alloc((size_t)GH * 4);
  int* ghs    = alloc((size_t)GH * 4);
  int* bs     = alloc(1024 * 4);
  int* cnts   = alloc(NBINS_F * 4);
  int* offs   = alloc(NBINS_F * 4);
  int* used   = alloc(NBINS_F * 4);
  int* cum    = alloc(NBINS_F * 4);
  int* flags  = alloc((size_t)n * 4);
  int* fscan  = alloc((size_t)n * 4);
  int* k0list = alloc((size_t)n * 4);
  int* bwi1k  = alloc((size_t)n * 4);
  int* lvlT   = alloc((size_t)n * 4);
  int* keepi  = alloc((size_t)n * 4);
  int* bwi0k  = alloc((size_t)n * 4);
  int* bwi1k2 = alloc((size_t)n * 4);
  int* lvl0kA = alloc((size_t)n * 4);
  int* lvl1kA = alloc((size_t)n * 4);
  int* f0i    = alloc((size_t)n * 4);
  int* f1i    = alloc((size_t)n * 4);
  int* keyF   = alloc((size_t)n * 4);

  k_init<<<1, 32, 0, stream>>>(scal, n, K);
  k_windows<<<NB, 256, 0, stream>>>(coors, n, bwi0, bwi1);

  // ---- phase 1: drop on unshifted windows (full set) ----
  k_prepsort<<<NB, 256, 0, stream>>>(bwi0, n, kA, iA);
  radix_sort16(kA, iA, kB, iB, &scal[0], gh, ghs, bs, NB, stream);
  k_zero<<<(NBINS_W + 255) / 256, 256, 0, stream>>>(cnts, NBINS_W);
  k_histfull<<<NB, 256, 0, stream>>>(bwi0, &scal[0], cnts);
  run_scan(cnts, offs, bs, NBINS_W, nullptr, stream);
  k_zero<<<NB, 256, 0, stream>>>(flags, n);
  k_inner_drop<<<NB, 256, 0, stream>>>(kA, iA, &scal[0], offs, cnts, flags, lvl0);
  run_scan(flags, fscan, bs, n, &scal[1], stream);          // total -> K0
  k_compact0<<<NB, 256, 0, stream>>>(flags, fscan, bwi1, n, k0list, bwi1k);

  // ---- phase 2: drop on shifted windows (K0 subset; bound read on device) ----
  k_prepsort<<<NB, 256, 0, stream>>>(bwi1k, n, kA, iA);
  radix_sort16(kA, iA, kB, iB, &scal[1], gh, ghs, bs, NB, stream);
  k_zero<<<(NBINS_W + 255) / 256, 256, 0, stream>>>(cnts, NBINS_W);
  k_histfull<<<NB, 256, 0, stream>>>(bwi1k, &scal[1], cnts);
  run_scan(cnts, offs, bs, NBINS_W, nullptr, stream);
  k_zero<<<NB, 256, 0, stream>>>(flags, n);
  k_inner_drop<<<NB, 256, 0, stream>>>(kA, iA, &scal[1], offs, cnts, flags, lvlT);
  run_scan(flags, fscan, bs, n, &scal[2], stream);          // total -> K (device)
  k_compact1<<<NB, 256, 0, stream>>>(flags, fscan, k0list, lvlT, &scal[1],
                                     keepi, lvl1kA);

  // ---- gather per-kept window ids / levels ----
  k_gather_kept<<<KB, 256, 0, stream>>>(keepi, bwi0, bwi1, lvl0, K,
                                        bwi0k, bwi1k2, lvl0kA);

  // ---- flat2win (both shifts), keyed by lvl*12800 + bwi ----
  const int* bwiS[2] = { bwi0k, bwi1k2 };
  const int* lvlS[2] = { lvl0kA, lvl1kA };
  int* fS[2] = { f0i, f1i };
  for (int s = 0; s < 2; ++s) {
    k_makekey<<<KB, 256, 0, stream>>>(bwiS[s], lvlS[s], K, keyF);
    k_prepsort<<<KB, 256, 0, stream>>>(keyF, K, kA, iA);
    radix_sort16(kA, iA, kB, iB, &scal[3], gh, ghs, bs, NB, stream);
    k_zero<<<(NBINS_F + 255) / 256, 256, 0, stream>>>(cnts, NBINS_F);
    k_histfull<<<KB, 256, 0, stream>>>(keyF, &scal[3], cnts);
    run_scan(cnts, offs, bs, NBINS_F, nullptr, stream);
    k_used<<<(NBINS_F + 255) / 256, 256, 0, stream>>>(cnts, NBINS_F, used);
    run_scan(used, cum, bs, NBINS_F, nullptr, stream);
    k_f2w<<<KB, 256, 0, stream>>>(kA, iA, K, offs, cum, fS[s]);
  }

  // ---- outputs (concatenated flat, float32) ----
  float* o = (float*)d_out;
  float* o_coors = o + (size_t)K * 128;
  float* o_keep  = o_coors + (size_t)K * 4;
  float* o_b0 = o_keep + K;
  float* o_b1 = o_b0 + K;
  float* o_l0 = o_b1 + K;
  float* o_l1 = o_l0 + K;
  float* o_f0 = o_l1 + K;
  float* o_f1 = o_f0 + K;
  float* o_p0 = o_f1 + K;
  float* o_p1 = o_p0 + (size_t)K * 128;

  k_out_feats<<<(K + 7) / 8, 256, 0, stream>>>(feats, keepi, o, K);
  k_out_misc<<<KB, 256, 0, stream>>>(coors, keepi, bwi0k, bwi1k2, lvl0kA, lvl1kA,
                                     f0i, f1i, o_coors, o_keep, o_b0, o_b1,
                                     o_l0, o_l1, o_f0, o_f1, K);
  k_pos<<<(K + 15) / 16, 32, 0, stream>>>(coors, keepi, o_p0, o_p1, K);
}